// LSTM_61194694034272
// MI455X (gfx1250) — compile-verified
//
#include <hip/hip_runtime.h>
#include <hip/hip_bf16.h>

typedef float v2f __attribute__((ext_vector_type(2)));
typedef float v8f __attribute__((ext_vector_type(8)));

static constexpr int kB = 256;    // batch
static constexpr int kT = 2048;   // time steps
static constexpr int kH = 64;     // hidden
static constexpr int kO = 12;     // out features
static constexpr int kL = 3;      // layers

// Single-instruction tanh (CDNA5 TRANS op v_tanh_f32).
__device__ __forceinline__ float tanh_fast(float x) {
    float r;
    asm("v_tanh_f32 %0, %1" : "=v"(r) : "v"(x));
    return r;
}
// sigmoid(x) = 0.5 + 0.5*tanh(x/2): 1 TRANS + 1 mul + 1 fma.
__device__ __forceinline__ float sigmoid_fast(float x) {
    return fmaf(0.5f, tanh_fast(0.5f * x), 0.5f);
}

// Relaxed workgroup barrier: orders LDS producers (s_wait_dscnt 0) and syncs,
// but deliberately does NOT drain LOADcnt/STOREcnt like __syncthreads() does —
// pipelined global loads/stores stay in flight across the sync.
__device__ __forceinline__ void barrier_lds() {
    asm volatile("s_wait_dscnt 0x0\n\t"
                 "s_barrier_signal -1\n\t"
                 "s_barrier_wait -1" ::: "memory");
}

// Swizzled row layout for x/h tiles (64 floats per row):
//   element j = 4*kt + koff  ->  (koff>>1)*32 + kt*2 + (koff&1)
// so a lane's WMMA A-tile reads (pairs {4kt+2*lhalf, +1}) are CONTIGUOUS across
// kt: addr = row*64 + lhalf*32 + kt*2  -> ds loads merge into b128.

// One LSTM layer. Each block owns a 16-row batch tile and runs the full T loop.
// 512 threads = 16 waves; wave w computes gate columns [16w, 16w+16) with two
// interleaved v_wmma_f32_16x16x4_f32 accumulation chains (Wih/Whh B-tiles in
// registers; A-tiles preloaded into register arrays so all ds_load_b128s issue
// ahead of the chain). x(t+1) is software-pipelined through registers (2x
// unrolled time loop), xls is double-buffered: 2 relaxed barriers per step.
// xin may alias hs_out (in-place across layers): every thread's x(t+1) load is
// consumed at the LDS commit before it signals barrier A(t+1); the overwriting
// stores to hs[t+1] happen only after barrier B(t+1). Tiles are block-disjoint.
template <int IN_DIM, bool BATCH_MAJOR>
__global__ __launch_bounds__(512) void lstm_layer_kernel(
    const float* xin,                  // BATCH_MAJOR: x[B][T][IN_DIM]; else hs[T][B][64]
    float* hs_out,                     // [T][B][64]
    const float* __restrict__ Wih,     // [256][IN_DIM]
    const float* __restrict__ Whh,     // [256][64]
    const float* __restrict__ bih,     // [256]
    const float* __restrict__ bhh,     // [256]
    const float* __restrict__ h0l,     // [B][64] (this layer's slice)
    const float* __restrict__ c0l,     // [B][64]
    float* __restrict__ hn_out,        // [B][64]
    float* __restrict__ cn_out)        // [B][64]
{
    constexpr int KT_IN = IN_DIM / 4;  // 15 (layer0) or 16

    __shared__ __align__(16) float xls[2][16 * 64];  // x tile (swizzled), double-buffered
    __shared__ __align__(16) float hls[16 * 64];     // h tile (swizzled)
    __shared__ __align__(16) float gls[16 * 256];    // gates [16][256] (row-major)

    const int tid  = threadIdx.x;
    const int lane = tid & 31;
    const int wave = tid >> 5;          // 0..15
    const int b0   = blockIdx.x * 16;

    const int lhalf = lane >> 4;        // 0: K{0,1}/C rows 0-7 ; 1: K{2,3}/C rows 8-15
    const int l15   = lane & 15;
    const int abase = l15 * 64 + lhalf * 32;  // A-tile read base (swizzled row)

    // ---- Preload WMMA B-tiles (weights) + bias into registers --------------
    // B 4x16 tile: lane L<16 -> VGPR{0,1}=B[k0+0..1][n0+L]; L>=16 -> B[k0+2..3][n0+L-16]
    // B[k][n] = W[n][k]  (gates = x @ W^T).
    const int n = wave * 16 + l15;      // this lane's gate column
    const float bias = bih[n] + bhh[n];

    v2f wih_b[KT_IN];
    v2f whh_b[16];
#pragma unroll
    for (int kt = 0; kt < KT_IN; ++kt)
        wih_b[kt] = *(const v2f*)(Wih + (size_t)n * IN_DIM + kt * 4 + lhalf * 2);
#pragma unroll
    for (int kt = 0; kt < 16; ++kt)
        whh_b[kt] = *(const v2f*)(Whh + (size_t)n * kH + kt * 4 + lhalf * 2);

    // ---- Initial h -> LDS (swizzled), c -> registers -----------------------
    {
        const int m = tid >> 5, p = tid & 31;            // source float2 pair p of row m
        const float2 h0v = ((const float2*)(h0l + (size_t)b0 * kH))[tid];
        *(float2*)&hls[m * 64 + (p & 1) * 32 + (p >> 1) * 2] = h0v;
    }

    const int b_e = tid >> 5;           // elementwise row 0..15
    const int j2  = (tid & 31) * 2;     // elementwise col (x2)
    const int hsw = b_e * 64 + ((j2 >> 1) & 1) * 32 + (j2 >> 2) * 2;  // swizzled h slot
    float2 cc = *(const float2*)(c0l + (size_t)(b0 + b_e) * kH + j2);
    float2 hh = make_float2(0.f, 0.f);

    // ---- x staging ownership -----------------------------------------------
    // Thread covers source float2 pair xp of row xm. BATCH_MAJOR rows have only
    // 30 pairs; 2 lanes duplicate the last pair -> no divergence, coalesced.
    const int xm = tid >> 5;
    const int xp = BATCH_MAJOR ? ((tid & 31) < 29 ? (tid & 31) : 29) : (tid & 31);
    const int xdst = xm * 64 + (xp & 1) * 32 + (xp >> 1) * 2;  // swizzled slot

    auto load_x = [&](int t) -> float2 {
        if (BATCH_MAJOR)
            return *(const float2*)(xin + ((size_t)(b0 + xm) * kT + t) * IN_DIM + xp * 2);
        else
            return *(const float2*)(xin + ((size_t)t * kB + b0) * kH + xm * 64 + xp * 2);
    };

    // One LSTM timestep. Commits xcur(t) to LDS, issues pipelined load of
    // x(t+1) into xnxt (consumed only by the NEXT body's commit).
    auto body = [&](int t, float2 xcur, float2& xnxt) {
        float* xbuf = xls[t & 1];
        *(float2*)&xbuf[xdst] = xcur;

        const int tn = (t + 1 < kT) ? (t + 1) : (kT - 1);  // clamped: no divergence
        xnxt = load_x(tn);

        barrier_lds();  // A: {prev elementwise hls writes, x commit} before WMMA reads

        // ---- Preload A-tiles into registers: distinct dests let the
        // scheduler issue all ds_load_b128s ahead of the WMMA chain. ---------
        v2f ax[KT_IN];
        v2f ah[16];
#pragma unroll
        for (int kt = 0; kt < KT_IN; ++kt)
            ax[kt] = *(const v2f*)&xbuf[abase + kt * 2];   // contiguous over kt
#pragma unroll
        for (int kt = 0; kt < 16; ++kt)
            ah[kt] = *(const v2f*)&hls[abase + kt * 2];    // contiguous over kt

        // ---- Gate GEMM: acc = bias + x@Wih^T + h@Whh^T  (fp32 WMMA) --------
        // Two interleaved accumulation chains to break the WMMA RAW chain.
        v8f acc0 = (v8f){bias, bias, bias, bias, bias, bias, bias, bias};
        v8f acc1 = (v8f){0.f, 0.f, 0.f, 0.f, 0.f, 0.f, 0.f, 0.f};

#pragma unroll
        for (int kt = 0; kt < KT_IN; ++kt) {
            if (kt & 1)
                acc1 = __builtin_amdgcn_wmma_f32_16x16x4_f32(
                    false, ax[kt], false, wih_b[kt], (short)0, acc1, false, false);
            else
                acc0 = __builtin_amdgcn_wmma_f32_16x16x4_f32(
                    false, ax[kt], false, wih_b[kt], (short)0, acc0, false, false);
        }
#pragma unroll
        for (int kt = 0; kt < 16; ++kt) {
            if (kt & 1)
                acc1 = __builtin_amdgcn_wmma_f32_16x16x4_f32(
                    false, ah[kt], false, whh_b[kt], (short)0, acc1, false, false);
            else
                acc0 = __builtin_amdgcn_wmma_f32_16x16x4_f32(
                    false, ah[kt], false, whh_b[kt], (short)0, acc0, false, false);
        }
        const v8f acc = acc0 + acc1;

        // ---- Scatter C tile to gate LDS ------------------------------------
        // C layout: lane L<16: VGPR r = C[r][L]; lane>=16: VGPR r = C[8+r][L-16]
#pragma unroll
        for (int r = 0; r < 8; ++r)
            gls[(r + 8 * lhalf) * 256 + n] = acc[r];

        barrier_lds();  // B: gate stores before elementwise reads

        // ---- Elementwise LSTM cell (c stays in registers) ------------------
        const float2 ig = *(const float2*)&gls[b_e * 256 + 0   + j2];
        const float2 fg = *(const float2*)&gls[b_e * 256 + 64  + j2];
        const float2 gg = *(const float2*)&gls[b_e * 256 + 128 + j2];
        const float2 og = *(const float2*)&gls[b_e * 256 + 192 + j2];

        cc.x = sigmoid_fast(fg.x) * cc.x + sigmoid_fast(ig.x) * tanh_fast(gg.x);
        cc.y = sigmoid_fast(fg.y) * cc.y + sigmoid_fast(ig.y) * tanh_fast(gg.y);
        hh.x = sigmoid_fast(og.x) * tanh_fast(cc.x);
        hh.y = sigmoid_fast(og.y) * tanh_fast(cc.y);

        *(float2*)&hls[hsw] = hh;                                        // swizzled
        *(float2*)(hs_out + ((size_t)t * kB + b0 + b_e) * kH + j2) = hh; // row-major
        // (no barrier here: barrier A of the next step orders hls/gls reuse)
    };

    float2 xA = load_x(0);
    float2 xB;
    barrier_lds();  // h0 staging visible before first WMMA
    for (int t = 0; t < kT; t += 2) {  // kT even; 2x unroll kills the xreg copy
        body(t, xA, xB);
        body(t + 1, xB, xA);
    }

    *(float2*)(hn_out + (size_t)(b0 + b_e) * kH + j2) = hh;
    *(float2*)(cn_out + (size_t)(b0 + b_e) * kH + j2) = cc;
}

// y[b][t][o] = hs[t][b][:] . W_out[o][:] + b_out[o]   (tiny, bandwidth-bound)
__global__ __launch_bounds__(256) void out_proj_kernel(
    const float* __restrict__ hs,    // [T][B][64]
    const float* __restrict__ Wout,  // [12][64]
    const float* __restrict__ bout,  // [12]
    float* __restrict__ y)           // [B][T][12]
{
    __shared__ float wls[kO * kH];
    __shared__ float bls[kO];
    const int tid = threadIdx.x;
    for (int i = tid; i < kO * kH; i += 256) wls[i] = Wout[i];
    if (tid < kO) bls[tid] = bout[tid];
    __syncthreads();

    const size_t g = (size_t)blockIdx.x * 256 + tid;  // g = t*B + b
    const int t = (int)(g >> 8);    // /256
    const int b = (int)(g & 255);   // %256
    const float* hrow = hs + g * kH;

    float acc[kO];
#pragma unroll
    for (int o = 0; o < kO; ++o) acc[o] = bls[o];
#pragma unroll
    for (int j4 = 0; j4 < kH; j4 += 4) {
        const float4 hv = *(const float4*)(hrow + j4);
#pragma unroll
        for (int o = 0; o < kO; ++o) {
            acc[o] += hv.x * wls[o * kH + j4 + 0];
            acc[o] += hv.y * wls[o * kH + j4 + 1];
            acc[o] += hv.z * wls[o * kH + j4 + 2];
            acc[o] += hv.w * wls[o * kH + j4 + 3];
        }
    }
    float4* yp = (float4*)(y + ((size_t)b * kT + t) * kO);  // 48B offsets: 16B-aligned
    yp[0] = make_float4(acc[0], acc[1], acc[2], acc[3]);
    yp[1] = make_float4(acc[4], acc[5], acc[6], acc[7]);
    yp[2] = make_float4(acc[8], acc[9], acc[10], acc[11]);
}

extern "C" void kernel_launch(void* const* d_in, const int* in_sizes, int n_in,
                              void* d_out, int out_size, void* d_ws, size_t ws_size,
                              hipStream_t stream) {
    const float* x    = (const float*)d_in[0];   // [B][T][60]
    const float* h0   = (const float*)d_in[1];   // [3][B][64]
    const float* c0   = (const float*)d_in[2];   // [3][B][64]
    const float* Wih[kL] = {(const float*)d_in[3], (const float*)d_in[7],  (const float*)d_in[11]};
    const float* Whh[kL] = {(const float*)d_in[4], (const float*)d_in[8],  (const float*)d_in[12]};
    const float* bih[kL] = {(const float*)d_in[5], (const float*)d_in[9],  (const float*)d_in[13]};
    const float* bhh[kL] = {(const float*)d_in[6], (const float*)d_in[10], (const float*)d_in[14]};
    const float* Wout = (const float*)d_in[15];  // [12][64]
    const float* bout = (const float*)d_in[16];  // [12]

    float* out = (float*)d_out;
    float* y   = out;                                   // B*T*12
    float* hn  = out + (size_t)kB * kT * kO;            // [3][B][64]
    float* cn  = hn + (size_t)kL * kB * kH;             // [3][B][64]

    float* hs = (float*)d_ws;  // [T][B][64] fp32 = 128 MB, reused in-place across layers

    const dim3 blk(512);       // 16 waves: one gate tile per wave
    const dim3 grd(kB / 16);   // 16 blocks, one per 16-row batch tile

    // Layer 0: batch-major x input -> hs
    lstm_layer_kernel<60, true><<<grd, blk, 0, stream>>>(
        x, hs, Wih[0], Whh[0], bih[0], bhh[0],
        h0 + 0 * (size_t)kB * kH, c0 + 0 * (size_t)kB * kH,
        hn + 0 * (size_t)kB * kH, cn + 0 * (size_t)kB * kH);

    // Layers 1,2: hs -> hs in place
    lstm_layer_kernel<64, false><<<grd, blk, 0, stream>>>(
        hs, hs, Wih[1], Whh[1], bih[1], bhh[1],
        h0 + 1 * (size_t)kB * kH, c0 + 1 * (size_t)kB * kH,
        hn + 1 * (size_t)kB * kH, cn + 1 * (size_t)kB * kH);

    lstm_layer_kernel<64, false><<<grd, blk, 0, stream>>>(
        hs, hs, Wih[2], Whh[2], bih[2], bhh[2],
        h0 + 2 * (size_t)kB * kH, c0 + 2 * (size_t)kB * kH,
        hn + 2 * (size_t)kB * kH, cn + 2 * (size_t)kB * kH);

    // Output projection
    out_proj_kernel<<<(kT * kB) / 256, dim3(256), 0, stream>>>(hs, Wout, bout, y);
}